// DUMFNet_73521250173308
// MI455X (gfx1250) — compile-verified
//
#include <hip/hip_runtime.h>
#include <math.h>

typedef __attribute__((ext_vector_type(2))) float v2f;
typedef __attribute__((ext_vector_type(8))) float v8f;

#define B_   32
#define CSUM 128

__device__ __forceinline__ float sig_(float x) { return 1.0f / (1.0f + expf(-x)); }

// ---------------------------------------------------------------- stats ----
// Per pixel (b,h,w): mean and max over channel dim. One thread per pixel,
// channel loop strided by HW so each iteration is a fully coalesced plane.
__global__ void stats_kernel(const float* __restrict__ t,
                             float* __restrict__ amean,
                             float* __restrict__ amax,
                             int C, int HW, int npix) {
    int pi = blockIdx.x * blockDim.x + threadIdx.x;
    if (pi >= npix) return;
    int b = pi / HW;
    int p = pi - b * HW;
    const float* base = t + (size_t)b * C * HW + p;
    float sum = 0.f;
    float mx  = -INFINITY;
    for (int c = 0; c < C; ++c) {
        float v = base[(size_t)c * HW];
        sum += v;
        mx = fmaxf(mx, v);
    }
    amean[pi] = sum / (float)C;
    amax[pi]  = mx;
}

// ------------------------------------------------------------- sp conv ----
// 7x7 kernel, dilation 3, pad 9 (same-size output), 2 input channels
// (mean, max), then sigmoid(. + sp_b).
__global__ void spconv_kernel(const float* __restrict__ amean,
                              const float* __restrict__ amax,
                              const float* __restrict__ sp_w,  // [2][7][7]
                              const float* __restrict__ sp_b,  // [1]
                              float* __restrict__ s_out,
                              int H, int W, int npix) {
    int pi = blockIdx.x * blockDim.x + threadIdx.x;
    if (pi >= npix) return;
    int HW = H * W;
    int b = pi / HW;
    int p = pi - b * HW;
    int h = p / W;
    int w = p - h * W;
    const float* am = amean + (size_t)b * HW;
    const float* ax = amax  + (size_t)b * HW;
    float acc = sp_b[0];
    #pragma unroll
    for (int kh = 0; kh < 7; ++kh) {
        int ih = h + kh * 3 - 9;
        if (ih < 0 || ih >= H) continue;
        #pragma unroll
        for (int kw = 0; kw < 7; ++kw) {
            int iw = w + kw * 3 - 9;
            if (iw < 0 || iw >= W) continue;
            int idx = ih * W + iw;
            acc += sp_w[kh * 7 + kw] * am[idx] + sp_w[49 + kh * 7 + kw] * ax[idx];
        }
    }
    s_out[pi] = sig_(acc);
}

// -------------------------------------------------------------- pooled ----
// pooled(b,c) = mean_hw t*(1+s). One block per (b,c); deterministic LDS
// tree reduction (no float atomics -> replay-stable results).
__global__ void pooled_kernel(const float* __restrict__ t,
                              const float* __restrict__ s_arr,
                              float* __restrict__ pooled,
                              int C, int HW, int c_off) {
    int bid = blockIdx.x;          // b*C + c
    int c = bid % C;
    int b = bid / C;
    const float* tb = t + ((size_t)b * C + c) * HW;
    const float* sb = s_arr + (size_t)b * HW;
    float sum = 0.f;
    for (int p = threadIdx.x; p < HW; p += blockDim.x)
        sum += tb[p] * (1.f + sb[p]);
    __shared__ float red[256];
    red[threadIdx.x] = sum;
    __syncthreads();
    for (int off = 128; off > 0; off >>= 1) {
        if ((int)threadIdx.x < off) red[threadIdx.x] += red[threadIdx.x + off];
        __syncthreads();
    }
    if (threadIdx.x == 0)
        pooled[b * CSUM + c_off + c] = red[0] / (float)HW;
}

// ---------------------------------------------------------------- catt ----
// Single block, 512 threads = 16 waves. Phase 1: conv1d(k=3,pad=1) over the
// channel dim of pooled -> att in LDS. Phase 2: catt = sigmoid(att @ W^T + b)
// as a 32x128x128 f32 GEMM using V_WMMA_F32_16X16X4_F32 (one 16x16 tile per
// wave, 32 k-steps of depth 4).
__global__ void __launch_bounds__(512)
catt_kernel(const float* __restrict__ pooled,
            const float* __restrict__ c1d_w,  // [3]
            const float* __restrict__ w1, const float* __restrict__ b1,
            const float* __restrict__ w2, const float* __restrict__ b2,
            const float* __restrict__ w3, const float* __restrict__ b3,
            const float* __restrict__ w4, const float* __restrict__ b4,
            const float* __restrict__ w5, const float* __restrict__ b5,
            float* __restrict__ catt) {
    __shared__ float att[B_][CSUM + 1];   // +1 pad: stride 129 dwords avoids bank conflicts
    int tid = threadIdx.x;

    float k0w = c1d_w[0], k1w = c1d_w[1], k2w = c1d_w[2];
    for (int i = tid; i < B_ * CSUM; i += 512) {
        int b = i >> 7;
        int k = i & 127;
        const float* pr = pooled + b * CSUM;
        float v = k1w * pr[k];
        if (k > 0)   v += k0w * pr[k - 1];
        if (k < 127) v += k2w * pr[k + 1];
        att[b][k] = v;
    }
    __syncthreads();

    int wave = tid >> 5;
    int lane = tid & 31;
    int tm   = wave >> 3;        // 0..1 (M tile)
    int tn   = wave & 7;         // 0..7 (N tile)
    int half = lane >> 4;        // 0: K=0,1   1: K=2,3
    int l16  = lane & 15;
    int m = tm * 16 + l16;       // A row this lane supplies
    int n = tn * 16 + l16;       // B column / output column this lane supplies

    // Output channel n maps into the stacked weight rows [w1;w2;w3;w4;w5].
    const float* wrow;
    float bias;
    if (n < 8)       { wrow = w1 + n * CSUM;        bias = b1[n]; }
    else if (n < 24) { wrow = w2 + (n - 8) * CSUM;  bias = b2[n - 8]; }
    else if (n < 48) { wrow = w3 + (n - 24) * CSUM; bias = b3[n - 24]; }
    else if (n < 80) { wrow = w4 + (n - 48) * CSUM; bias = b4[n - 48]; }
    else             { wrow = w5 + (n - 80) * CSUM; bias = b5[n - 80]; }

    // Accumulator column is fixed per lane -> bias broadcast into all 8 rows.
    v8f acc;
    #pragma unroll
    for (int i = 0; i < 8; ++i) acc[i] = bias;

    int koff = half * 2;
    for (int k0 = 0; k0 < CSUM; k0 += 4) {
        v2f a, bf;
        a.x  = att[m][k0 + koff];
        a.y  = att[m][k0 + koff + 1];
        bf.x = wrow[k0 + koff];
        bf.y = wrow[k0 + koff + 1];
        acc = __builtin_amdgcn_wmma_f32_16x16x4_f32(
            /*neg_a=*/false, a, /*neg_b=*/false, bf,
            /*c_mod=*/(short)0, acc, /*reuse_a=*/false, /*reuse_b=*/false);
    }

    #pragma unroll
    for (int i = 0; i < 8; ++i) {
        int mm = tm * 16 + half * 8 + i;   // D row per C/D VGPR layout
        catt[mm * CSUM + n] = sig_(acc[i]);
    }
}

// ----------------------------------------------------------------- out ----
// out = t * (catt*(1+s) + s); fully coalesced on t/out, s broadcast over c.
__global__ void out_kernel(const float* __restrict__ t,
                           const float* __restrict__ s_arr,
                           const float* __restrict__ catt,
                           float* __restrict__ out,
                           int C, int HW, int c_off, int n) {
    int idx = blockIdx.x * blockDim.x + threadIdx.x;
    if (idx >= n) return;
    int chw = C * HW;
    int b = idx / chw;
    int rem = idx - b * chw;
    int c = rem / HW;
    int p = rem - c * HW;
    float sv = s_arr[b * HW + p];
    float cv = catt[b * CSUM + c_off + c];
    out[idx] = t[idx] * (cv * (1.f + sv) + sv);
}

// -------------------------------------------------------------- launch ----
extern "C" void kernel_launch(void* const* d_in, const int* in_sizes, int n_in,
                              void* d_out, int out_size, void* d_ws, size_t ws_size,
                              hipStream_t stream) {
    const float* t[5];
    for (int i = 0; i < 5; ++i) t[i] = (const float*)d_in[i];
    const float* sp_w  = (const float*)d_in[5];
    const float* sp_b  = (const float*)d_in[6];
    const float* c1d_w = (const float*)d_in[7];
    const float* w[5];
    const float* bb[5];
    for (int i = 0; i < 5; ++i) {
        w[i]  = (const float*)d_in[8 + 2 * i];
        bb[i] = (const float*)d_in[9 + 2 * i];
    }

    static const int Cs[5]   = {8, 16, 24, 32, 48};
    static const int Hs[5]   = {256, 128, 64, 32, 16};
    static const int coff[5] = {0, 8, 24, 48, 80};

    // pixel counts & offsets
    size_t poff[5], ooff[5];
    size_t pacc = 0, oacc = 0;
    int npix[5], nelm[5];
    for (int i = 0; i < 5; ++i) {
        int HW = Hs[i] * Hs[i];
        npix[i] = B_ * HW;
        nelm[i] = B_ * Cs[i] * HW;
        poff[i] = pacc; pacc += (size_t)npix[i];
        ooff[i] = oacc; oacc += (size_t)nelm[i];
    }
    const size_t NP = pacc;  // 2,793,472

    float* ws     = (float*)d_ws;
    float* amean  = ws;
    float* amax   = ws + NP;
    float* sarr   = ws + 2 * NP;
    float* pooled = ws + 3 * NP;
    float* catt   = pooled + B_ * CSUM;
    float* outp   = (float*)d_out;

    const int TB = 256;
    for (int i = 0; i < 5; ++i) {
        int HW = Hs[i] * Hs[i];
        int gp = (npix[i] + TB - 1) / TB;
        stats_kernel<<<gp, TB, 0, stream>>>(t[i], amean + poff[i], amax + poff[i],
                                            Cs[i], HW, npix[i]);
        spconv_kernel<<<gp, TB, 0, stream>>>(amean + poff[i], amax + poff[i],
                                             sp_w, sp_b, sarr + poff[i],
                                             Hs[i], Hs[i], npix[i]);
        pooled_kernel<<<B_ * Cs[i], TB, 0, stream>>>(t[i], sarr + poff[i], pooled,
                                                     Cs[i], HW, coff[i]);
    }

    catt_kernel<<<1, 512, 0, stream>>>(pooled, c1d_w,
                                       w[0], bb[0], w[1], bb[1], w[2], bb[2],
                                       w[3], bb[3], w[4], bb[4], catt);

    for (int i = 0; i < 5; ++i) {
        int HW = Hs[i] * Hs[i];
        int go = (nelm[i] + TB - 1) / TB;
        out_kernel<<<go, TB, 0, stream>>>(t[i], sarr + poff[i], catt,
                                          outp + ooff[i], Cs[i], HW, coff[i], nelm[i]);
    }
}